// AttentionLayer_50294067036847
// MI455X (gfx1250) — compile-verified
//
#include <hip/hip_runtime.h>
#include <stdint.h>
#include <math.h>

#define BATCH 64
#define TLEN  4096
#define DDIM  256
#define UDIM  256
#define TT    64              // t-rows per block in score kernel

typedef float v2f __attribute__((ext_vector_type(2)));
typedef float v8f __attribute__((ext_vector_type(8)));

// gfx1250 has a hardware V_TANH_F32 ("tanh-insts"); use it if the builtin is
// exposed, otherwise fall back to libm tanhf (safe: compile-time guarded).
#if __has_builtin(__builtin_amdgcn_tanhf)
__device__ __forceinline__ float fast_tanh(float x) {
  return __builtin_amdgcn_tanhf(x);
}
#elif __has_builtin(__builtin_amdgcn_tanh_f32)
__device__ __forceinline__ float fast_tanh(float x) {
  return __builtin_amdgcn_tanh_f32(x);
}
#else
__device__ __forceinline__ float fast_tanh(float x) { return tanhf(x); }
#endif

// ---------------------------------------------------------------------------
// Kernel 1: P[b,u] = query[b,:]@W1[:,u] + b1[u] + b2[u]   (both biases folded)
// ---------------------------------------------------------------------------
__global__ __launch_bounds__(256) void qproj_kernel(
    const float* __restrict__ query, const float* __restrict__ W1,
    const float* __restrict__ b1, const float* __restrict__ b2,
    float* __restrict__ P) {
  const int b = blockIdx.x;
  const int u = threadIdx.x;
  const float* q = query + b * DDIM;
  float acc = b1[u] + b2[u];
#pragma unroll 4
  for (int d = 0; d < DDIM; ++d)
    acc = fmaf(q[d], W1[d * UDIM + u], acc);
  P[b * UDIM + u] = acc;
}

// ---------------------------------------------------------------------------
// Kernel 2: score[b,t] = tanh(P[b,:] + values[b,t,:]@W2) @ V + bV
// One block per (b, 64-row t-tile). 8 waves; wave w owns u-tiles 2w, 2w+1
// and ALL four 16-row t-subtiles -> each W2 fragment feeds 4 WMMAs.
// A (16x4 f32) from LDS (filled by async-to-LDS), B from W2 (L2-resident),
// fp32 WMMA accumulate over K=256 in steps of 4.
// ---------------------------------------------------------------------------
__global__ __launch_bounds__(256) void score_kernel(
    const float* __restrict__ values, const float* __restrict__ W2,
    const float* __restrict__ P, const float* __restrict__ V,
    const float* __restrict__ bV, float* __restrict__ score) {
  const int b    = blockIdx.y;
  const int t0   = blockIdx.x * TT;
  const int tid  = threadIdx.x;
  const int lane = tid & 31;
  const int wave = tid >> 5;
  const int half = lane >> 4;   // 0: lanes 0-15, 1: lanes 16-31
  const int l16  = lane & 15;

  __shared__ __align__(16) float Atile[TT][260];  // 64 t-rows x 256 k (+pad)
  __shared__ float score_s[TT];

  if (tid < TT) score_s[tid] = bV[0];

  // Async copy of the 64x256 fp32 values tile into LDS (CDNA5 async path).
  // 4096 float4 chunks, 16 per thread; row stride padded to 260 floats (16B
  // aligned) to kill DS bank conflicts on the A-fragment reads.
  const float* gbase = values + ((size_t)b * TLEN + t0) * DDIM;
#pragma unroll
  for (int it = 0; it < 16; ++it) {
    const int f   = tid + it * 256;   // float4 index 0..4095
    const int row = f >> 6;           // 64 float4 per row
    const int c4  = f & 63;
    unsigned lds_off = (unsigned)(uintptr_t)(&Atile[row][c4 * 4]);
    unsigned goff    = (unsigned)((row * DDIM + c4 * 4) * sizeof(float));
    asm volatile("global_load_async_to_lds_b128 %0, %1, %2"
                 :: "v"(lds_off), "v"(goff), "s"(gbase)
                 : "memory");
  }
  asm volatile("s_wait_asynccnt 0" ::: "memory");
  __syncthreads();

  const int u0a = (wave * 2 + 0) * 16;
  const int u0b = (wave * 2 + 1) * 16;

  v8f acc0[4] = {v8f{}, v8f{}, v8f{}, v8f{}};
  v8f acc1[4] = {v8f{}, v8f{}, v8f{}, v8f{}};

  // K loop: V_WMMA_F32_16X16X4_F32, 64 steps of K=4, 8 WMMA per step.
  // A layout (ISA 16x4 MxK): lane = M (both halves); VGPR0 = K {0|2},
  // VGPR1 = K {1|3}. B layout (4x16 KxN, transpose-symmetric): lane = N.
  for (int k0 = 0; k0 < DDIM; k0 += 4) {
    const int ka = k0 + half * 2;

    const float* w2p = W2 + (size_t)ka * UDIM;
    v2f bfa, bfb;
    bfa.x = w2p[u0a + l16];
    bfa.y = w2p[UDIM + u0a + l16];
    bfb.x = w2p[u0b + l16];
    bfb.y = w2p[UDIM + u0b + l16];

    v2f a[4];
#pragma unroll
    for (int tt = 0; tt < 4; ++tt) {
      a[tt].x = Atile[tt * 16 + l16][ka];
      a[tt].y = Atile[tt * 16 + l16][ka + 1];
    }

#pragma unroll
    for (int tt = 0; tt < 4; ++tt) {
      acc0[tt] = __builtin_amdgcn_wmma_f32_16x16x4_f32(
          false, a[tt], false, bfa, (short)0, acc0[tt], false, false);
      acc1[tt] = __builtin_amdgcn_wmma_f32_16x16x4_f32(
          false, a[tt], false, bfb, (short)0, acc1[tt], false, false);
    }
  }

  // Epilogue: x = acc + P[b,u]; s += tanh(x)*V[u]; reduce over N (16 lanes
  // per half, matching C/D layout: VGPR i holds M=i (half 0) / M=i+8 (half 1)).
  const int u_a = u0a + l16, u_b = u0b + l16;
  const float pa = P[b * UDIM + u_a], pb = P[b * UDIM + u_b];
  const float va = V[u_a],            vb = V[u_b];
#pragma unroll
  for (int tt = 0; tt < 4; ++tt) {
#pragma unroll
    for (int i = 0; i < 8; ++i) {
      float s = fast_tanh(acc0[tt][i] + pa) * va +
                fast_tanh(acc1[tt][i] + pb) * vb;
      s += __shfl_xor(s, 1);
      s += __shfl_xor(s, 2);
      s += __shfl_xor(s, 4);
      s += __shfl_xor(s, 8);               // stays within each 16-lane half
      if (l16 == 0)
        atomicAdd(&score_s[tt * 16 + i + half * 8], s);  // ds_add_f32
    }
  }
  __syncthreads();
  if (tid < TT) score[(size_t)b * TLEN + t0 + tid] = score_s[tid];
}

// ---------------------------------------------------------------------------
// Kernel 3: softmax over T per batch; also zero-init context row (d_out is
// poisoned by the harness).
// ---------------------------------------------------------------------------
__global__ __launch_bounds__(256) void softmax_kernel(
    const float* __restrict__ score, float* __restrict__ attn,
    float* __restrict__ ctx) {
  const int b = blockIdx.x, tid = threadIdx.x;
  __shared__ float red[256];
  const float* s = score + (size_t)b * TLEN;
  float* a = attn + (size_t)b * TLEN;

  float lmax = -INFINITY;
  for (int i = tid; i < TLEN; i += 256) lmax = fmaxf(lmax, s[i]);
  red[tid] = lmax;
  __syncthreads();
  for (int off = 128; off > 0; off >>= 1) {
    if (tid < off) red[tid] = fmaxf(red[tid], red[tid + off]);
    __syncthreads();
  }
  const float gmax = red[0];
  __syncthreads();

  float lsum = 0.f;
  for (int i = tid; i < TLEN; i += 256) {
    const float e = __expf(s[i] - gmax);
    a[i] = e;
    lsum += e;
  }
  red[tid] = lsum;
  __syncthreads();
  for (int off = 128; off > 0; off >>= 1) {
    if (tid < off) red[tid] += red[tid + off];
    __syncthreads();
  }
  const float inv = 1.0f / red[0];
  for (int i = tid; i < TLEN; i += 256) a[i] *= inv;

  ctx[b * DDIM + tid] = 0.f;  // D == 256 == blockDim
}

// ---------------------------------------------------------------------------
// Kernel 4: context[b,d] = sum_t attn[b,t] * values[b,t,d]  (split-T partials
// combined with global_atomic_add_f32)
// ---------------------------------------------------------------------------
__global__ __launch_bounds__(256) void context_kernel(
    const float* __restrict__ values, const float* __restrict__ attn,
    float* __restrict__ ctx) {
  const int b  = blockIdx.y;
  const int t0 = blockIdx.x * 512;
  const int d  = threadIdx.x;
  const float* vp = values + ((size_t)b * TLEN + t0) * DDIM + d;
  const float* ap = attn + (size_t)b * TLEN + t0;
  float acc = 0.f;
  for (int t = 0; t < 512; ++t)
    acc = fmaf(ap[t], vp[(size_t)t * DDIM], acc);
  atomicAdd(&ctx[b * DDIM + d], acc);
}

// ---------------------------------------------------------------------------
extern "C" void kernel_launch(void* const* d_in, const int* in_sizes, int n_in,
                              void* d_out, int out_size, void* d_ws,
                              size_t ws_size, hipStream_t stream) {
  const float* query  = (const float*)d_in[0];
  const float* values = (const float*)d_in[1];
  const float* W1     = (const float*)d_in[2];
  const float* b1     = (const float*)d_in[3];
  const float* W2     = (const float*)d_in[4];
  const float* b2     = (const float*)d_in[5];
  const float* V      = (const float*)d_in[6];
  const float* bV     = (const float*)d_in[7];

  float* ctx  = (float*)d_out;              // [B, D]
  float* attn = ctx + BATCH * DDIM;         // [B, T, 1]

  float* P     = (float*)d_ws;              // [B, U]
  float* score = P + BATCH * UDIM;          // [B, T]

  qproj_kernel<<<BATCH, 256, 0, stream>>>(query, W1, b1, b2, P);
  score_kernel<<<dim3(TLEN / TT, BATCH), 256, 0, stream>>>(values, W2, P, V,
                                                           bV, score);
  softmax_kernel<<<BATCH, 256, 0, stream>>>(score, attn, ctx);
  context_kernel<<<dim3(TLEN / 512, BATCH), 256, 0, stream>>>(values, attn,
                                                              ctx);
}